// AdaFace_32169305047284
// MI455X (gfx1250) — compile-verified
//
#include <hip/hip_runtime.h>
#include <math.h>

#define M_MARGIN 0.4f
#define EPS_F    1e-3f
#define S_SCALE  64.0f
#define PI_F     3.14159265358979323846f

typedef float v2f __attribute__((ext_vector_type(2)));
typedef float v4f __attribute__((ext_vector_type(4)));
typedef float v8f __attribute__((ext_vector_type(8)));

// ---------------------------------------------------------------------------
// Wave32 all-lane sum using V_WMMA_F32_16X16X4_F32.
// A (16x4 f32): lane L<16 holds A[L][0]=a.x, A[L][1]=a.y; lane L>=16 holds
// A[L-16][2]=a.x, A[L-16][3]=a.y.  With a.y = 0 and B = all-ones:
//   D[m][n] = psum(m) + psum(m+16)   (same for every column n)
// D (16x16 f32, 8 VGPRs): lane n<16 holds D[r][n] in d[r]; lane n+16 holds
// D[r+8][n].  Summing d[0..7] then xor-16 shuffle yields the full 32-lane sum
// in every lane.
// ---------------------------------------------------------------------------
__device__ __forceinline__ float wave_sum_wmma(float v) {
    v2f a; a.x = v;    a.y = 0.0f;
    v2f b; b.x = 1.0f; b.y = 1.0f;
    v8f c = {};
    v8f d = __builtin_amdgcn_wmma_f32_16x16x4_f32(
        /*neg_a=*/false, a, /*neg_b=*/false, b,
        /*c_mod=*/(short)0, c, /*reuse_a=*/false, /*reuse_b=*/false);
    float s = d[0] + d[1] + d[2] + d[3] + d[4] + d[5] + d[6] + d[7];
    s += __shfl_xor(s, 16, 32);   // combine M=0..7 half with M=8..15 half
    return s;                     // total in every lane
}

// ---------------------------------------------------------------------------
// Kernel 1: mean / std of clipped norms over positive rows (one wave).
// ---------------------------------------------------------------------------
__global__ void adaface_stats_kernel(const float* __restrict__ norms,
                                     const int*   __restrict__ labels,
                                     int Bn, float* __restrict__ stats) {
    const int lane = threadIdx.x;   // blockDim.x == 32, single wave
    float psum = 0.0f, pcnt = 0.0f;
    for (int i = lane; i < Bn; i += 32) {
        float sn  = fminf(fmaxf(norms[i], 1e-3f), 100.0f);
        bool  pos = (labels[i] != -1);
        psum += pos ? sn   : 0.0f;
        pcnt += pos ? 1.0f : 0.0f;
    }
    float total = wave_sum_wmma(psum);
    float cnt   = wave_sum_wmma(pcnt);
    float mean  = total / cnt;

    float vsum = 0.0f;
    for (int i = lane; i < Bn; i += 32) {
        float sn  = fminf(fmaxf(norms[i], 1e-3f), 100.0f);
        bool  pos = (labels[i] != -1);
        float dv  = sn - mean;
        vsum += pos ? dv * dv : 0.0f;
    }
    float var  = wave_sum_wmma(vsum) / (cnt - 1.0f);   // unbiased, torch.std
    float stdv = sqrtf(var);
    if (lane == 0) { stats[0] = mean; stats[1] = stdv; }
}

// ---------------------------------------------------------------------------
// Kernel 2: streaming elementwise transform.  row = blockIdx.y so the
// positive-row branch is uniform per block (no divergence).  float4 (B128)
// non-temporal loads/stores: the ~800MB stream should not thrash L2.
// ---------------------------------------------------------------------------
__global__ void adaface_transform_kernel(const float* __restrict__ logits,
                                         const int*   __restrict__ labels,
                                         float* __restrict__ out, int C) {
    const int row  = blockIdx.y;
    const int c4   = blockIdx.x * blockDim.x + threadIdx.x;
    const int nvec = C >> 2;
    if (c4 >= nvec) return;

    const bool pos = (labels[row] != -1);
    const v4f* src = (const v4f*)(logits + (size_t)row * C);
    v4f x = __builtin_nontemporal_load(src + c4);
    v4f r;
    if (pos) {
#pragma unroll
        for (int j = 0; j < 4; ++j) {
            float th = acosf(x[j]);
            th = fminf(fmaxf(th, EPS_F), PI_F - EPS_F);
            r[j] = cosf(th) * S_SCALE;
        }
    } else {
        r = x * S_SCALE;
    }
    v4f* dst = (v4f*)(out + (size_t)row * C);
    __builtin_nontemporal_store(r, dst + c4);
}

// Scalar tail (only launched if C % 4 != 0; C=100000 -> not launched).
__global__ void adaface_tail_kernel(const float* __restrict__ logits,
                                    const int*   __restrict__ labels,
                                    float* __restrict__ out, int C, int tail) {
    const int row = blockIdx.x;
    const int t   = threadIdx.x;
    if (t >= tail) return;
    const size_t idx = (size_t)row * C + (C - tail) + t;
    const bool pos = (labels[row] != -1);
    float x = logits[idx];
    float r;
    if (pos) {
        float th = fminf(fmaxf(acosf(x), EPS_F), PI_F - EPS_F);
        r = cosf(th);
    } else {
        r = x;
    }
    out[idx] = r * S_SCALE;
}

// ---------------------------------------------------------------------------
// Kernel 3: per-row fixup at (row, label): angular + additive margin.
// ---------------------------------------------------------------------------
__global__ void adaface_fixup_kernel(const float* __restrict__ logits,
                                     const float* __restrict__ norms,
                                     const int*   __restrict__ labels,
                                     const float* __restrict__ stats,
                                     float* __restrict__ out, int Bn, int C) {
    const int row = blockIdx.x * blockDim.x + threadIdx.x;
    if (row >= Bn) return;
    const int lab = labels[row];
    if (lab < 0) return;

    const float mean = stats[0];
    const float stdv = stats[1];
    const float sn   = fminf(fmaxf(norms[row], 1e-3f), 100.0f);
    const float ms   = (sn - mean) / (stdv + EPS_F);
    const float g_angular = -M_MARGIN * ms;
    const float g_add     =  M_MARGIN + M_MARGIN * ms;

    const size_t idx = (size_t)row * C + (size_t)lab;
    float th = acosf(logits[idx]) + g_angular;
    th = fminf(fmaxf(th, EPS_F), PI_F - EPS_F);
    out[idx] = (cosf(th) - g_add) * S_SCALE;
}

// ---------------------------------------------------------------------------
extern "C" void kernel_launch(void* const* d_in, const int* in_sizes, int n_in,
                              void* d_out, int out_size, void* d_ws, size_t ws_size,
                              hipStream_t stream) {
    const float* logits = (const float*)d_in[0];
    const float* norms  = (const float*)d_in[1];
    const int*   labels = (const int*)d_in[2];
    float*       out    = (float*)d_out;
    float*       stats  = (float*)d_ws;       // [0]=mean, [1]=std

    const int Bn = in_sizes[1];               // norms is [B,1] -> B elements
    const int C  = in_sizes[0] / Bn;          // logits is [B,C]

    // 1) tiny stats reduction (one wave; WMMA cross-lane sums)
    adaface_stats_kernel<<<1, 32, 0, stream>>>(norms, labels, Bn, stats);

    // 2) memory-bound streaming transform, float4 per thread
    const int nvec = C / 4;
    dim3 grid((nvec + 255) / 256, Bn);
    adaface_transform_kernel<<<grid, 256, 0, stream>>>(logits, labels, out, C);

    const int tail = C % 4;
    if (tail) {
        adaface_tail_kernel<<<Bn, 32, 0, stream>>>(logits, labels, out, C, tail);
    }

    // 3) one element per positive row gets the margins
    adaface_fixup_kernel<<<(Bn + 255) / 256, 256, 0, stream>>>(
        logits, norms, labels, stats, out, Bn, C);
}